// KvPageState_16621523436393
// MI455X (gfx1250) — compile-verified
//
#include <hip/hip_runtime.h>
#include <stdint.h>

// ---------------------------------------------------------------------------
// KV page-state scatter for MI455X (gfx1250).
//   out = copy(kv_pages); out[dest_slot] = concat(new_k[t], new_v[t]) for valid t
// Pure memory-movement op: optimize via NT cache hints, async global<->LDS
// (ASYNCcnt) bulk copy, and skipping destination slots that will be fully
// overwritten (bitmap in d_ws).
// ---------------------------------------------------------------------------

typedef float v4f __attribute__((ext_vector_type(4)));

#define SLOT_FLOATS 2048                 // 2 * KV_HEADS(8) * HEAD_DIM(128)
#define SLOT_V4     (SLOT_FLOATS / 4)    // 512 x float4 = 8 KB per slot
#define TOK_V4      (SLOT_V4 / 2)        // 256 x float4 per token per tensor (K or V)

// --- K0a: zero the slot bitmap ---------------------------------------------
__global__ void kv_zero_bitmap(uint32_t* __restrict__ bm, int nwords) {
    int i = blockIdx.x * blockDim.x + threadIdx.x;
    if (i < nwords) bm[i] = 0u;
}

// --- K0b: mark slots that will be fully overwritten ------------------------
__global__ void kv_mark_dests(const int* __restrict__ dests, int ntok,
                              uint32_t* __restrict__ bm) {
    int t = blockIdx.x * blockDim.x + threadIdx.x;
    if (t >= ntok) return;
    int d = dests[t];
    if (d >= 0) atomicOr(&bm[d >> 5], 1u << (d & 31));
}

// --- K1: bulk copy kv_pages -> out via async global<->LDS, skipping marked
//         slots. One 8KB slot region per block-iteration; 256 threads, each
//         lane moves 2 x b128. Double-buffered LDS -> single asynccnt wait
//         per region; multi-wave occupancy hides the per-wave stalls.
__global__ void kv_copy_pages_async(const v4f* __restrict__ src,
                                    v4f* __restrict__ dst,
                                    const uint32_t* __restrict__ bm,
                                    int nslots) {
    __shared__ v4f lds[2][SLOT_V4];      // 2 x 8 KB double buffer
    const unsigned tid = threadIdx.x;
    int p = 0;
    for (int r = blockIdx.x; r < nslots; r += gridDim.x) {
        if (bm) {
            uint32_t w = bm[r >> 5];               // uniform per block
            if ((w >> (r & 31)) & 1u) continue;    // slot fully overwritten later
        }
        const v4f* s0 = src + (size_t)r * SLOT_V4 + tid;
        const v4f* s1 = s0 + 256;
        v4f*       d0 = dst + (size_t)r * SLOT_V4 + tid;
        v4f*       d1 = d0 + 256;
        unsigned l0 = (unsigned)(uintptr_t)&lds[p][tid];        // LDS byte addr
        unsigned l1 = (unsigned)(uintptr_t)&lds[p][tid + 256];

        // global -> LDS (async, non-temporal read: 512MB stream won't fit L2)
        asm volatile("global_load_async_to_lds_b128 %0, %1, off th:TH_LOAD_NT"
                     :: "v"(l0), "v"(s0) : "memory");
        asm volatile("global_load_async_to_lds_b128 %0, %1, off th:TH_LOAD_NT"
                     :: "v"(l1), "v"(s1) : "memory");
        // retire our loads (and the previous region's stores -> LDS reuse safe)
        asm volatile("s_wait_asynccnt 0" ::: "memory");
        // LDS -> global (async, non-temporal write)
        asm volatile("global_store_async_from_lds_b128 %0, %1, off th:TH_STORE_NT"
                     :: "v"(d0), "v"(l0) : "memory");
        asm volatile("global_store_async_from_lds_b128 %0, %1, off th:TH_STORE_NT"
                     :: "v"(d1), "v"(l1) : "memory");
        p ^= 1;   // stores drain while next region loads into the other buffer
    }
    // s_endpgm performs an implicit wait-idle, so trailing stores complete.
}

// --- K2: scatter new_k/new_v into their slot regions ------------------------
// One block per token; 8 KB contiguous destination; fully coalesced b128 NT.
__global__ void kv_scatter(const v4f* __restrict__ k4,
                           const v4f* __restrict__ v4,
                           const int* __restrict__ dests,
                           v4f* __restrict__ dst) {
    const int t = blockIdx.x;
    const int d = dests[t];
    if (d < 0) return;                       // padded token: dropped
    const unsigned tid = threadIdx.x;
    const size_t ob = (size_t)d * SLOT_V4;
    const size_t ib = (size_t)t * TOK_V4;
    v4f kv = __builtin_nontemporal_load(&k4[ib + tid]);
    v4f vv = __builtin_nontemporal_load(&v4[ib + tid]);
    __builtin_nontemporal_store(kv, &dst[ob + tid]);          // heads 0..7
    __builtin_nontemporal_store(vv, &dst[ob + 256 + tid]);    // heads 8..15
}

// ---------------------------------------------------------------------------
extern "C" void kernel_launch(void* const* d_in, const int* in_sizes, int n_in,
                              void* d_out, int out_size, void* d_ws, size_t ws_size,
                              hipStream_t stream) {
    const float* kv_pages = (const float*)d_in[0];
    const float* new_k    = (const float*)d_in[1];
    const float* new_v    = (const float*)d_in[2];
    const int*   dests    = (const int*)d_in[3];
    float*       out      = (float*)d_out;

    const int total  = in_sizes[0];            // 134,217,728 floats (512 MB)
    const int nslots = total / SLOT_FLOATS;    // 65,536 slot regions
    const int ntok   = in_sizes[3];            // 8,192 tokens
    const int nwords = (nslots + 31) / 32;     // 2,048 bitmap words (8 KB)

    uint32_t* bm = nullptr;
    if (ws_size >= (size_t)nwords * sizeof(uint32_t)) {
        bm = (uint32_t*)d_ws;
        kv_zero_bitmap<<<(nwords + 255) / 256, 256, 0, stream>>>(bm, nwords);
        kv_mark_dests<<<(ntok + 255) / 256, 256, 0, stream>>>(dests, ntok, bm);
    }

    int copy_blocks = nslots < 8192 ? nslots : 8192;   // 8 regions per block
    kv_copy_pages_async<<<copy_blocks, 256, 0, stream>>>(
        (const v4f*)kv_pages, (v4f*)out, bm, nslots);

    kv_scatter<<<ntok, 256, 0, stream>>>(
        (const v4f*)new_k, (const v4f*)new_v, dests, (v4f*)out);
}